// GraphAttnSfMProjectionFeatureUpdate_33088428049090
// MI455X (gfx1250) — compile-verified
//
#include <hip/hip_runtime.h>

typedef __attribute__((ext_vector_type(2))) float v2f;
typedef __attribute__((ext_vector_type(8))) float v8f;

#define LN_EPS 1e-5f

__device__ __forceinline__ float wave_reduce_add(float v) {
#pragma unroll
  for (int off = 16; off > 0; off >>= 1)
    v += __shfl_xor(v, off, 32);
  return v;
}

// ---------------------------------------------------------------------------
// Kernel 1: global row -> bias2 = b_proj + relu(LN(global)) @ W_g.T
// Single block of 128 threads.
// ---------------------------------------------------------------------------
__global__ void __launch_bounds__(128)
global_bias_kernel(const float* __restrict__ gf,
                   const float* __restrict__ gg, const float* __restrict__ gb,
                   const float* __restrict__ Wg, const float* __restrict__ bp,
                   float* __restrict__ bias2) {
  __shared__ float red[128];
  __shared__ float gln[256];
  const int t = threadIdx.x;
  const float x0 = gf[t];
  const float x1 = gf[t + 128];

  // mean
  red[t] = x0 + x1;
  __syncthreads();
#pragma unroll
  for (int off = 64; off > 0; off >>= 1) {
    if (t < off) red[t] += red[t + off];
    __syncthreads();
  }
  const float mean = red[0] * (1.f / 256.f);
  __syncthreads();

  // variance (E[x^2] - mean^2)
  red[t] = x0 * x0 + x1 * x1;
  __syncthreads();
#pragma unroll
  for (int off = 64; off > 0; off >>= 1) {
    if (t < off) red[t] += red[t + off];
    __syncthreads();
  }
  const float var = red[0] * (1.f / 256.f) - mean * mean;
  const float rstd = rsqrtf(var + LN_EPS);

  gln[t]       = fmaxf((x0 - mean) * rstd * gg[t]       + gb[t],       0.f);
  gln[t + 128] = fmaxf((x1 - mean) * rstd * gg[t + 128] + gb[t + 128], 0.f);
  __syncthreads();

  float acc = bp[t];
  const float* wrow = Wg + (size_t)t * 256;
#pragma unroll 8
  for (int k = 0; k < 256; ++k) acc = fmaf(gln[k], wrow[k], acc);
  bias2[t] = acc;
}

// ---------------------------------------------------------------------------
// Kernel 2: node tables.  Out[n,128] = relu(LN(X[n,256])) @ W.T, W:[128,256]
// One wave per 16-row tile; LN via wave shuffles into padded LDS tile,
// then f32 WMMA 16x16x4 over K=256 (64 steps) for 8 column tiles.
// ---------------------------------------------------------------------------
__global__ void __launch_bounds__(128)
node_project_kernel(const float* __restrict__ X,
                    const float* __restrict__ lng, const float* __restrict__ lnb,
                    const float* __restrict__ W, float* __restrict__ Out, int n) {
  __shared__ float lds[4 * 16 * 258];  // 4 waves * 16 rows * padded 258
  const int lane = threadIdx.x & 31;
  // force wave-uniform values into SGPRs -> scalar branches, no exec churn
  const int warp = __builtin_amdgcn_readfirstlane(threadIdx.x >> 5);
  const int ntiles = (n + 15) >> 4;
  const int tile = blockIdx.x * 4 + warp;
  if (tile >= ntiles) return;

  float* A = lds + warp * (16 * 258);
  const int m16 = lane & 15;
  const int h = lane >> 4;
  const int c0 = lane * 8;
  const bool full = (tile * 16 + 16 <= n);  // scalar

  float g8[8], b8[8];
#pragma unroll
  for (int t = 0; t < 8; ++t) { g8[t] = lng[c0 + t]; b8[t] = lnb[c0 + t]; }

  // LayerNorm + ReLU: 16 rows, each reduced wave-wide (8 elems / lane)
  for (int r = 0; r < 16; ++r) {
    int row = tile * 16 + r;
    if (row >= n) row = n - 1;
    const float* x = X + (size_t)row * 256;
    const float4 v0 = *(const float4*)(x + c0);
    const float4 v1 = *(const float4*)(x + c0 + 4);
    const float xv[8] = {v0.x, v0.y, v0.z, v0.w, v1.x, v1.y, v1.z, v1.w};
    float s = 0.f, q = 0.f;
#pragma unroll
    for (int t = 0; t < 8; ++t) { s += xv[t]; q += xv[t] * xv[t]; }
    s = wave_reduce_add(s);
    q = wave_reduce_add(q);
    const float mean = s * (1.f / 256.f);
    const float var = q * (1.f / 256.f) - mean * mean;
    const float rstd = rsqrtf(var + LN_EPS);
    float* arow = A + r * 258;
#pragma unroll
    for (int t = 0; t < 8; ++t) {
      const float y = (xv[t] - mean) * rstd * g8[t] + b8[t];
      arow[c0 + t] = fmaxf(y, 0.f);
    }
  }

  // GEMM: [16x256] @ [256x128], f32 WMMA 16x16x4
  const float* aarow = A + m16 * 258 + 2 * h;  // A-frag base for this lane
  for (int nt = 0; nt < 8; ++nt) {
    v8f acc = {};
    const float* wrow = W + (size_t)(nt * 16 + m16) * 256 + 2 * h;  // B-frag base
#pragma unroll 8
    for (int kb = 0; kb < 64; ++kb) {
      const v2f af = *(const v2f*)(aarow + kb * 4);
      const v2f bf = *(const v2f*)(wrow + kb * 4);
      acc = __builtin_amdgcn_wmma_f32_16x16x4_f32(false, af, false, bf,
                                                  (short)0, acc, false, false);
    }
    const int col = nt * 16 + m16;
    float* obase = Out + (size_t)(tile * 16 + 8 * h) * 128 + col;
    if (full) {
      // hot path: straight-line stores (scalar branch guards this block)
#pragma unroll
      for (int j = 0; j < 8; ++j) obase[(size_t)j * 128] = acc[j];
    } else {
#pragma unroll
      for (int j = 0; j < 8; ++j) {
        const int row = tile * 16 + 8 * h + j;
        if (row < n) obase[(size_t)j * 128] = acc[j];
      }
    }
  }
}

// ---------------------------------------------------------------------------
// Kernel 3: fused main kernel over nnz rows.
// out[i,:] = 0.25*(proj[i,:] @ Wp.T + bias2 + sp_p[pt_idx[i],:] + vw_p[view_idx[i],:])
// W_proj staged in LDS (padded stride 130); one wave per 16-row tile,
// A-fragments resident in 64 VGPRs, 8 N-tiles x 32 K-steps of f32 WMMA.
// ---------------------------------------------------------------------------
__global__ void __launch_bounds__(128)
fused_proj_kernel(const float* __restrict__ proj,
                  const int* __restrict__ vidx, const int* __restrict__ pidx,
                  const float* __restrict__ Wp, const float* __restrict__ bias2,
                  const float* __restrict__ sp_p, const float* __restrict__ vw_p,
                  float* __restrict__ out, int nnz) {
  __shared__ float ldsW[128 * 130];
  // cooperatively stage W_proj (row-major [n][k], padded row stride 130)
  for (int i = threadIdx.x; i < 128 * 128; i += 128)
    ldsW[(i >> 7) * 130 + (i & 127)] = Wp[i];
  __syncthreads();

  const int lane = threadIdx.x & 31;
  // force wave-uniform values into SGPRs -> scalar branches, no exec churn
  const int warp = __builtin_amdgcn_readfirstlane(threadIdx.x >> 5);
  const int m16 = lane & 15;
  const int h = lane >> 4;
  const int ntiles = (nnz + 15) >> 4;
  const int totalWaves = gridDim.x * 4;

  // bias2 preloaded once per wave: lane's column in each of the 8 N-tiles
  float bias2v[8];
#pragma unroll
  for (int nt = 0; nt < 8; ++nt) bias2v[nt] = bias2[nt * 16 + m16];

  for (int tile = blockIdx.x * 4 + warp; tile < ntiles; tile += totalWaves) {
    const int r0 = tile * 16;
    const bool full = (r0 + 16 <= nnz);  // scalar

    // A fragments: lane holds row (r0+m16), K-pair (4*kb + 2h, +1)
    int ra = r0 + m16;
    if (!full && ra > nnz - 1) ra = nnz - 1;
    const float* arow = proj + (size_t)ra * 128 + 2 * h;
    v2f a[32];
#pragma unroll
    for (int kb = 0; kb < 32; ++kb) a[kb] = *(const v2f*)(arow + kb * 4);

    // gathered row base pointers for the 8 output rows this lane touches
    const float* sprow[8];
    const float* vwrow[8];
#pragma unroll
    for (int j = 0; j < 8; ++j) {
      int rr = r0 + 8 * h + j;
      if (!full && rr > nnz - 1) rr = nnz - 1;
      sprow[j] = sp_p + (size_t)pidx[rr] * 128;
      vwrow[j] = vw_p + (size_t)vidx[rr] * 128;
    }

    for (int ntile = 0; ntile < 8; ++ntile) {
      v8f acc = {};
      const float* wbase = ldsW + (ntile * 16 + m16) * 130 + 2 * h;
#pragma unroll
      for (int kb = 0; kb < 32; ++kb) {
        const v2f bf = *(const v2f*)(wbase + kb * 4);
        acc = __builtin_amdgcn_wmma_f32_16x16x4_f32(false, a[kb], false, bf,
                                                    (short)0, acc, false, false);
      }
      const int col = ntile * 16 + m16;
      const float bv = bias2v[ntile];
      float* obase = out + (size_t)(r0 + 8 * h) * 128 + col;
      if (full) {
        // hot path: straight-line loads + stores, scalar-branch guarded
#pragma unroll
        for (int j = 0; j < 8; ++j) {
          const float o = (acc[j] + bv + sprow[j][col] + vwrow[j][col]) * 0.25f;
          obase[(size_t)j * 128] = o;
        }
      } else {
#pragma unroll
        for (int j = 0; j < 8; ++j) {
          const int rr = r0 + 8 * h + j;
          const float o = (acc[j] + bv + sprow[j][col] + vwrow[j][col]) * 0.25f;
          if (rr < nnz) obase[(size_t)j * 128] = o;
        }
      }
    }
  }
}

// ---------------------------------------------------------------------------
extern "C" void kernel_launch(void* const* d_in, const int* in_sizes, int n_in,
                              void* d_out, int out_size, void* d_ws, size_t ws_size,
                              hipStream_t stream) {
  const float* sp_feat  = (const float*)d_in[0];
  const float* vw_feat  = (const float*)d_in[1];
  const float* gl_feat  = (const float*)d_in[2];
  const float* proj_v   = (const float*)d_in[3];
  const int*   view_idx = (const int*)d_in[4];
  const int*   pt_idx   = (const int*)d_in[5];
  const float* ln_sp_g  = (const float*)d_in[6];
  const float* ln_sp_b  = (const float*)d_in[7];
  const float* ln_vw_g  = (const float*)d_in[8];
  const float* ln_vw_b  = (const float*)d_in[9];
  const float* ln_g_g   = (const float*)d_in[10];
  const float* ln_g_b   = (const float*)d_in[11];
  const float* W_proj   = (const float*)d_in[12];
  const float* b_proj   = (const float*)d_in[13];
  const float* W_sp     = (const float*)d_in[14];
  const float* W_view   = (const float*)d_in[15];
  const float* W_g      = (const float*)d_in[16];
  float* out = (float*)d_out;

  const int n_pts   = in_sizes[0] / 256;
  const int n_views = in_sizes[1] / 256;
  const int nnz     = in_sizes[4];

  // workspace layout (floats): [bias2:128][vw_p:n_views*128][sp_p:n_pts*128]
  float* bias2 = (float*)d_ws;
  float* vw_p  = bias2 + 128;
  float* sp_p  = vw_p + (size_t)n_views * 128;

  global_bias_kernel<<<1, 128, 0, stream>>>(gl_feat, ln_g_g, ln_g_b, W_g, b_proj, bias2);

  const int vt = (n_views + 15) / 16;
  node_project_kernel<<<(vt + 3) / 4, 128, 0, stream>>>(vw_feat, ln_vw_g, ln_vw_b,
                                                        W_view, vw_p, n_views);

  const int st = (n_pts + 15) / 16;
  node_project_kernel<<<(st + 3) / 4, 128, 0, stream>>>(sp_feat, ln_sp_g, ln_sp_b,
                                                        W_sp, sp_p, n_pts);

  const int nt = (nnz + 15) / 16;
  int blocks = (nt + 3) / 4;
  if (blocks > 4096) blocks = 4096;
  fused_proj_kernel<<<blocks, 128, 0, stream>>>(proj_v, view_idx, pt_idx, W_proj,
                                                bias2, sp_p, vw_p, out, nnz);
}